// GPTTransformer_12481174962814
// MI455X (gfx1250) — compile-verified
//
#include <hip/hip_runtime.h>
#include <hip/hip_bf16.h>
#include <math.h>

// ---------------- model dims ----------------
#define BB   8
#define TT   512
#define EE   256
#define DD   8
#define HH   8
#define LL   16
#define FFD  1024
#define VV   50257
#define MM   (BB*TT)      // 4096 token rows

// ---------------- GEMM tiling ----------------
#define TILE_M 64
#define TILE_N 128
#define TILE_K 16
#define GEMM_THREADS 256

#define F_ACC  1
#define F_RELU 2

typedef float v2f __attribute__((ext_vector_type(2)));
typedef float v8f __attribute__((ext_vector_type(8)));
typedef int   v4i __attribute__((ext_vector_type(4)));

#define GLOBAL_AS __attribute__((address_space(1)))
#define LDS_AS    __attribute__((address_space(3)))

#if defined(__HIP_DEVICE_COMPILE__) && \
    __has_builtin(__builtin_amdgcn_global_load_async_to_lds_b32) && \
    __has_builtin(__builtin_amdgcn_global_load_async_to_lds_b128)
#define HAVE_ASYNC_LDS 1
#else
#define HAVE_ASYNC_LDS 0
#endif

// ---------------------------------------------------------------------------
// Generic fp32 WMMA GEMM:  C[M,N] (op)= A[M,K] @ B[K,N] (+bias) (+relu)
// 256 threads = 8 waves; block tile 64x128; each wave owns 4 16x16 C tiles.
// Uses V_WMMA_F32_16X16X4_F32 (native CDNA5 fp32 matrix path).
// B tile stored interleaved [k/2][n][k&1] so each WMMA B operand is one
// aligned 8-byte LDS load into an even VGPR pair (no packing movs).
// Full-N blocks stream tiles with GLOBAL_LOAD_ASYNC_TO_LDS (ASYNCcnt,
// double-buffered); partial-N blocks use a guarded synchronous path.
// Requires: M % 64 == 0, K % 16 == 0, A rows 16B-aligned. N arbitrary.
// ---------------------------------------------------------------------------
__global__ __launch_bounds__(GEMM_THREADS)
void wmma_gemm_f32(const float* __restrict__ A, const float* __restrict__ B,
                   float* __restrict__ C, const float* __restrict__ bias,
                   int M, int N, int K, int flags)
{
    __shared__ alignas(16) float sA[2][TILE_M * TILE_K];  // 64x16 row-major
    __shared__ alignas(16) float sB[2][TILE_K * TILE_N];  // interleaved [k/2][n][k&1]

    const int tid  = threadIdx.x;
    const int lane = tid & 31;
    const int wave = tid >> 5;
    const int rt   = wave >> 1;            // row tile 0..3 (16 rows each)
    const int cg   = wave & 1;             // col group 0..1 (64 cols each)
    const int m0   = blockIdx.y * TILE_M;
    const int n0   = blockIdx.x * TILE_N;
    const int lm   = lane & 15;
    const int koff = (lane >> 4) << 1;     // 0 or 2

    v8f acc0 = {}; v8f acc1 = {}; v8f acc2 = {}; v8f acc3 = {};

    const int nk = K / TILE_K;
    const bool fullN = (n0 + TILE_N <= N);

    // per-thread load coordinates
    const int ar  = tid >> 2;              // A row 0..63
    const int akq = (tid & 3) << 2;        // A k offset 0,4,8,12

    auto load_sync = [&](int k0, int buf) {
        *(float4*)&sA[buf][ar * TILE_K + akq] =
            *(const float4*)&A[(size_t)(m0 + ar) * K + k0 + akq];
        #pragma unroll
        for (int i = 0; i < (TILE_K * TILE_N) / GEMM_THREADS; ++i) {
            int idx = tid + i * GEMM_THREADS;
            int kk = idx >> 7, nn = idx & 127;
            int gn = n0 + nn;
            sB[buf][((kk >> 1) * TILE_N + nn) * 2 + (kk & 1)] =
                (gn < N) ? B[(size_t)(k0 + kk) * N + gn] : 0.0f;
        }
    };

    auto compute = [&](int buf) {
        #pragma unroll
        for (int kk = 0; kk < TILE_K; kk += 4) {
            v2f a = *(const v2f*)&sA[buf][(rt * 16 + lm) * TILE_K + kk + koff];
            const int rr = (kk + koff) >> 1;     // even row pair index
            const int nb = cg * 64 + lm;
            v2f b0 = *(const v2f*)&sB[buf][(rr * TILE_N + nb +  0) * 2];
            v2f b1 = *(const v2f*)&sB[buf][(rr * TILE_N + nb + 16) * 2];
            v2f b2 = *(const v2f*)&sB[buf][(rr * TILE_N + nb + 32) * 2];
            v2f b3 = *(const v2f*)&sB[buf][(rr * TILE_N + nb + 48) * 2];
            acc0 = __builtin_amdgcn_wmma_f32_16x16x4_f32(false, a, false, b0, (short)0, acc0, false, false);
            acc1 = __builtin_amdgcn_wmma_f32_16x16x4_f32(false, a, false, b1, (short)0, acc1, false, false);
            acc2 = __builtin_amdgcn_wmma_f32_16x16x4_f32(false, a, false, b2, (short)0, acc2, false, false);
            acc3 = __builtin_amdgcn_wmma_f32_16x16x4_f32(false, a, false, b3, (short)0, acc3, false, false);
        }
    };

#if HAVE_ASYNC_LDS
    auto load_async = [&](int k0, int buf) {
        // A: one B128 per thread (16B aligned: K%16==0, rows 16B aligned)
        __builtin_amdgcn_global_load_async_to_lds_b128(
            (GLOBAL_AS v4i*)(A + (size_t)(m0 + ar) * K + k0 + akq),
            (LDS_AS v4i*)&sA[buf][ar * TILE_K + akq], 0, 0);
        // B: eight B32 per thread, scattered into interleaved layout
        #pragma unroll
        for (int i = 0; i < (TILE_K * TILE_N) / GEMM_THREADS; ++i) {
            int idx = tid + i * GEMM_THREADS;
            int kk = idx >> 7, nn = idx & 127;
            __builtin_amdgcn_global_load_async_to_lds_b32(
                (GLOBAL_AS int*)(B + (size_t)(k0 + kk) * N + n0 + nn),
                (LDS_AS int*)&sB[buf][((kk >> 1) * TILE_N + nn) * 2 + (kk & 1)], 0, 0);
        }
    };

    if (fullN) {
        // double-buffered async pipeline: 9 async instrs per wave per tile
        load_async(0, 0);
        for (int s = 0; s < nk; ++s) {
            if (s + 1 < nk) {
                load_async((s + 1) * TILE_K, (s + 1) & 1);
                asm volatile("s_wait_asynccnt 0x9" ::: "memory");
            } else {
                asm volatile("s_wait_asynccnt 0x0" ::: "memory");
            }
            __syncthreads();
            compute(s & 1);
            __syncthreads();
        }
    } else
#endif
    {
        for (int s = 0; s < nk; ++s) {
            load_sync(s * TILE_K, 0);
            __syncthreads();
            compute(0);
            __syncthreads();
        }
    }

    // epilogue: bias / residual-accumulate / relu, N-guarded
    const int rbase = m0 + rt * 16 + ((lane >> 4) << 3);
    v8f accs[4] = {acc0, acc1, acc2, acc3};
    #pragma unroll
    for (int ct = 0; ct < 4; ++ct) {
        int gn = n0 + cg * 64 + ct * 16 + lm;
        if (gn < N) {
            float bval = bias ? bias[gn] : 0.0f;
            #pragma unroll
            for (int r = 0; r < 8; ++r) {
                size_t off = (size_t)(rbase + r) * N + gn;
                float v = accs[ct][r] + bval;
                if (flags & F_ACC)  v += C[off];
                if (flags & F_RELU) v = fmaxf(v, 0.0f);
                C[off] = v;
            }
        }
    }
}

// ---------------------------------------------------------------------------
// x[m,:] = tok_emb[tokens[m],:] + pos_emb[m % T,:]
// ---------------------------------------------------------------------------
__global__ __launch_bounds__(256)
void embed_kernel(const int* __restrict__ tokens, const float* __restrict__ tok_emb,
                  const float* __restrict__ pos_emb, float* __restrict__ x)
{
    int m = blockIdx.x, e = threadIdx.x;
    int tok = tokens[m];
    int t = m & (TT - 1);
    x[(size_t)m * EE + e] = tok_emb[(size_t)tok * EE + e] + pos_emb[(size_t)t * EE + e];
}

// ---------------------------------------------------------------------------
// LayerNorm over E=256 (one block per row, 256 threads)
// ---------------------------------------------------------------------------
__global__ __launch_bounds__(256)
void ln_kernel(const float* __restrict__ x, const float* __restrict__ s,
               const float* __restrict__ b, float* __restrict__ out)
{
    __shared__ float red[256];
    int m = blockIdx.x, t = threadIdx.x;
    float v = x[(size_t)m * EE + t];
    red[t] = v; __syncthreads();
    for (int st = 128; st > 0; st >>= 1) { if (t < st) red[t] += red[t + st]; __syncthreads(); }
    float mean = red[0] * (1.0f / EE);
    __syncthreads();
    float d = v - mean;
    red[t] = d * d; __syncthreads();
    for (int st = 128; st > 0; st >>= 1) { if (t < st) red[t] += red[t + st]; __syncthreads(); }
    float var = red[0] * (1.0f / EE);
    out[(size_t)m * EE + t] = d * rsqrtf(var + 1e-5f) * s[t] + b[t];
}

// ---------------------------------------------------------------------------
// Repack Wq/Wk/Wv (L,H,E,D) into row-major [L][E][3*H*D] for a single QKV GEMM
// ---------------------------------------------------------------------------
__global__ __launch_bounds__(256)
void repack_qkv_kernel(const float* __restrict__ Wq, const float* __restrict__ Wk,
                       const float* __restrict__ Wv, float* __restrict__ wqkv)
{
    int idx = blockIdx.x * 256 + threadIdx.x;      // L*E*192 total
    int n = idx % 192;
    int e = (idx / 192) & (EE - 1);
    int l = idx / (192 * EE);
    int h = (n & 63) >> 3, d = n & 7;
    const float* src = (n < 64) ? Wq : (n < 128) ? Wk : Wv;
    wqkv[idx] = src[(((size_t)l * HH + h) * EE + e) * DD + d];
}

// ---------------------------------------------------------------------------
// R[b,h,i,j] = softmax_j( mask_triu( sum_t K[b,h,t,i]*Q[b,h,t,j] / 16 ) )
// qkv layout per row m: [Q(64) | K(64) | V(64)], one block per (b,h).
// ---------------------------------------------------------------------------
__global__ __launch_bounds__(256)
void attn_r_kernel(const float* __restrict__ qkv, float* __restrict__ R)
{
    __shared__ float part[256];
    __shared__ float Rm[64];
    __shared__ float Em[64];
    int bh = blockIdx.x;
    int b = bh >> 3, h = bh & 7;
    int tid = threadIdx.x;
    int g = tid >> 6, p = tid & 63;
    int i = p >> 3, j = p & 7;
    float s = 0.0f;
    int tstart = g * (TT / 4), tend = tstart + (TT / 4);
    for (int t = tstart; t < tend; ++t) {
        size_t base = (size_t)(b * TT + t) * 192;
        s += qkv[base + 64 + h * 8 + i] * qkv[base + h * 8 + j];
    }
    part[tid] = s;
    __syncthreads();
    if (tid < 64) Rm[p] = (part[p] + part[64 + p] + part[128 + p] + part[192 + p]) * (1.0f / 16.0f);
    __syncthreads();
    float e = 0.0f;
    if (tid < 64) {
        float mx = -3.0e38f;
        for (int jj = i; jj < 8; ++jj) mx = fmaxf(mx, Rm[i * 8 + jj]);
        e = (j >= i) ? expf(Rm[p] - mx) : 0.0f;
        Em[p] = e;
    }
    __syncthreads();
    if (tid < 64) {
        float sum = 0.0f;
        for (int jj = i; jj < 8; ++jj) sum += Em[i * 8 + jj];
        R[(size_t)bh * 64 + p] = e / sum;
    }
}

// ---------------------------------------------------------------------------
// a[m, h*8+j] = sum_i V[m, h*8+i] * R[b,h,i,j]   (output already [B,T,H*D])
// ---------------------------------------------------------------------------
__global__ __launch_bounds__(256)
void attn_av_kernel(const float* __restrict__ qkv, const float* __restrict__ R,
                    float* __restrict__ a)
{
    int idx = blockIdx.x * 256 + threadIdx.x;  // MM*64 total
    int m = idx >> 6, c = idx & 63;
    int h = c >> 3, j = c & 7;
    int b = m >> 9;                            // m / TT
    const float* Rp = R + (size_t)(b * HH + h) * 64;
    size_t base = (size_t)m * 192 + 128 + h * 8;
    float s = 0.0f;
    #pragma unroll
    for (int i2 = 0; i2 < 8; ++i2) s += qkv[base + i2] * Rp[i2 * 8 + j];
    a[(size_t)m * 64 + c] = s;
}

// ---------------------------------------------------------------------------
// Per-row cross-entropy: rowloss[m] = logsumexp(logits[m,:]) - logits[m,tgt]
// ---------------------------------------------------------------------------
__global__ __launch_bounds__(256)
void loss_row_kernel(const float* __restrict__ logits, const int* __restrict__ tgt,
                     float* __restrict__ rowloss)
{
    __shared__ float red[256];
    int m = blockIdx.x, t = threadIdx.x;
    const float* row = logits + (size_t)m * VV;
    float mx = -3.0e38f;
    for (int n = t; n < VV; n += 256) mx = fmaxf(mx, row[n]);
    red[t] = mx; __syncthreads();
    for (int st = 128; st > 0; st >>= 1) { if (t < st) red[t] = fmaxf(red[t], red[t + st]); __syncthreads(); }
    float M_ = red[0]; __syncthreads();
    float s = 0.0f;
    for (int n = t; n < VV; n += 256) s += expf(row[n] - M_);
    red[t] = s; __syncthreads();
    for (int st = 128; st > 0; st >>= 1) { if (t < st) red[t] += red[t + st]; __syncthreads(); }
    if (t == 0) rowloss[m] = logf(red[0]) + M_ - row[tgt[m]];
}

__global__ __launch_bounds__(256)
void loss_final_kernel(const float* __restrict__ rowloss, float* __restrict__ out)
{
    __shared__ float red[256];
    int t = threadIdx.x;
    float s = 0.0f;
    for (int m = t; m < MM; m += 256) s += rowloss[m];
    red[t] = s; __syncthreads();
    for (int st = 128; st > 0; st >>= 1) { if (t < st) red[t] += red[t + st]; __syncthreads(); }
    if (t == 0) out[(size_t)MM * VV] = red[0] * (1.0f / MM);
}

// ---------------------------------------------------------------------------
extern "C" void kernel_launch(void* const* d_in, const int* in_sizes, int n_in,
                              void* d_out, int out_size, void* d_ws, size_t ws_size,
                              hipStream_t stream)
{
    (void)in_sizes; (void)n_in; (void)out_size; (void)ws_size;

    const int*   tokens  = (const int*)d_in[0];
    const int*   targets = (const int*)d_in[1];
    const float* tok_emb = (const float*)d_in[2];
    const float* pos_emb = (const float*)d_in[3];
    const float* Wq      = (const float*)d_in[4];
    const float* Wk      = (const float*)d_in[5];
    const float* Wv      = (const float*)d_in[6];
    const float* Wo      = (const float*)d_in[7];
    const float* bo      = (const float*)d_in[8];
    const float* ln1_s   = (const float*)d_in[9];
    const float* ln1_b   = (const float*)d_in[10];
    const float* W1      = (const float*)d_in[11];
    const float* b1      = (const float*)d_in[12];
    const float* W2      = (const float*)d_in[13];
    const float* b2      = (const float*)d_in[14];
    const float* ln2_s   = (const float*)d_in[15];
    const float* ln2_b   = (const float*)d_in[16];
    const float* lnf_s   = (const float*)d_in[17];
    const float* lnf_b   = (const float*)d_in[18];
    const float* Wf      = (const float*)d_in[19];
    const float* bf      = (const float*)d_in[20];

    float* out = (float*)d_out;

    // workspace carve-out (~33 MB of floats)
    float* ws = (float*)d_ws;
    size_t o = 0;
    float* x       = ws + o; o += (size_t)MM * EE;       // activations
    float* hbuf    = ws + o; o += (size_t)MM * EE;       // LN output
    float* qkv     = ws + o; o += (size_t)MM * 192;      // fused QKV
    float* ubuf    = ws + o; o += (size_t)MM * FFD;      // MLP hidden
    float* abuf    = ws + o; o += (size_t)MM * 64;       // attention out (pre-proj)
    float* Rbuf    = ws + o; o += (size_t)BB * HH * 64;  // 8x8 attn matrices
    float* wqkv    = ws + o; o += (size_t)LL * EE * 192; // packed QKV weights
    float* rowloss = ws + o; o += (size_t)MM;

    dim3 blk(256);

    repack_qkv_kernel<<<(LL * EE * 192) / 256, blk, 0, stream>>>(Wq, Wk, Wv, wqkv);
    embed_kernel<<<MM, blk, 0, stream>>>(tokens, tok_emb, pos_emb, x);

    auto gemm = [&](const float* A, const float* Bm, float* C, const float* bias,
                    int M, int N, int K, int flags) {
        dim3 grid((N + TILE_N - 1) / TILE_N, M / TILE_M);
        wmma_gemm_f32<<<grid, blk, 0, stream>>>(A, Bm, C, bias, M, N, K, flags);
    };

    for (int l = 0; l < LL; ++l) {
        ln_kernel<<<MM, blk, 0, stream>>>(x, ln1_s + l * EE, ln1_b + l * EE, hbuf);
        gemm(hbuf, wqkv + (size_t)l * EE * 192, qkv, nullptr, MM, 192, EE, 0);
        attn_r_kernel<<<BB * HH, blk, 0, stream>>>(qkv, Rbuf);
        attn_av_kernel<<<(MM * 64) / 256, blk, 0, stream>>>(qkv, Rbuf, abuf);
        gemm(abuf, Wo + (size_t)l * 64 * EE, x, bo + l * EE, MM, EE, 64, F_ACC);
        ln_kernel<<<MM, blk, 0, stream>>>(x, ln2_s + l * EE, ln2_b + l * EE, hbuf);
        gemm(hbuf, W1 + (size_t)l * EE * FFD, ubuf, b1 + l * FFD, MM, FFD, EE, F_RELU);
        gemm(ubuf, W2 + (size_t)l * FFD * EE, x, b2 + l * EE, MM, EE, FFD, F_ACC);
    }

    ln_kernel<<<MM, blk, 0, stream>>>(x, lnf_s, lnf_b, hbuf);
    gemm(hbuf, Wf, out, bf, MM, VV, EE, 0);          // 105 GFLOP vocab GEMM -> d_out
    loss_row_kernel<<<MM, blk, 0, stream>>>(out, targets, rowloss);
    loss_final_kernel<<<1, blk, 0, stream>>>(rowloss, out);
}